// GatingNetwork_57062935494767
// MI455X (gfx1250) — compile-verified
//
#include <hip/hip_runtime.h>
#include <math.h>

// GatingNetwork fused kernel for MI455X (gfx1250, wave32).
//
//   h = relu(x @ W1 + b1); logit = mean(h @ W2 + b2)
//   alpha = softmax([logit1, logit2]); out = a0*f1 + a1*f2
//
// Memory-bound (~768 MB HBM traffic -> ~33us @ 23.3 TB/s). Fused design:
//   prep kernel:  w1t[16][4096] = zero-padded transpose of W1 (in d_ws)
//   main kernel:  per 16-row tile, x@W1 via V_WMMA_F32_16X16X4_F32 with
//                 K split across wave pairs; epilogue + combine in-block.

typedef float v2f __attribute__((ext_vector_type(2)));
typedef float v8f __attribute__((ext_vector_type(8)));

#define B_ROWS   16384
#define D_DIM    4096
#define H_DIM    7
#define OUT_DIM  7
#define N_PAD    16        // WMMA N dimension (cols 7..15 zero)

#define WAVES_PER_BLOCK 8
#define BLOCK_THREADS   (WAVES_PER_BLOCK * 32)            // 256
#define ROWS_PER_WAVE   16
#define TILES_PER_BLOCK (WAVES_PER_BLOCK / 2)             // 2 waves per tile (K-split)
#define ROWS_PER_BLOCK  (TILES_PER_BLOCK * ROWS_PER_WAVE) // 64
#define K_HALF          (D_DIM / 2)                       // 2048

__device__ __forceinline__ v2f lo2(const float4 v) { v2f r; r.x = v.x; r.y = v.y; return r; }
__device__ __forceinline__ v2f hi2(const float4 v) { v2f r; r.x = v.z; r.y = v.w; return r; }

// ---- prep: w1t[n][k] = (n < 7) ? W1[k][n] : 0  (zero-padded transpose) ----
__global__ __launch_bounds__(256)
void prep_w1t_kernel(const float* __restrict__ W1, float* __restrict__ w1t) {
    const int idx = blockIdx.x * 256 + threadIdx.x;   // 0 .. 16*4096-1
    const int n = idx >> 12;
    const int k = idx & (D_DIM - 1);
    w1t[idx] = (n < H_DIM) ? W1[(size_t)k * H_DIM + n] : 0.0f;
}

__global__ __launch_bounds__(BLOCK_THREADS)
void gating_fused_kernel(const float* __restrict__ f1,
                         const float* __restrict__ f2,
                         const float* __restrict__ w1t,  // [16][4096] padded W1^T
                         const float* __restrict__ b1,   // [7]
                         const float* __restrict__ W2,   // [7,7]
                         const float* __restrict__ b2,   // [7]
                         float* __restrict__ out)        // [B, D]
{
    // Per-wave partial C staging (cols 0..6, padded to 8) + per-row alphas.
    __shared__ float ldsC[WAVES_PER_BLOCK][2][ROWS_PER_WAVE][8];
    __shared__ float ldsAlpha[ROWS_PER_BLOCK][2];

    const int tid   = threadIdx.x;
    const int wave  = tid >> 5;
    const int lane  = tid & 31;
    const int tile  = wave >> 1;       // which 16-row tile in the block
    const int khalf = wave & 1;        // which half of K this wave reduces

    const int blockRow = blockIdx.x * ROWS_PER_BLOCK;
    const int waveRow  = blockRow + tile * ROWS_PER_WAVE;

    // WMMA_F32_16X16X4_F32 fragments (wave32):
    //   A: lane L, vgpr v <- x[M=L&15][kmap(v)]     B: lane L, vgpr v <- W1T[N=L&15][kmap(v)]
    //   C: lane L, vgpr r <- [M = r + 8*(L>>4)][N = L&15]
    // K remap: one float4 at (k + 4*(L>>4)) feeds two chunks:
    //   chunk0 = {k,k+1, k+4,k+5} (xy), chunk1 = {k+2,k+3, k+6,k+7} (zw).
    const int mA    = lane & 15;
    const int nB    = lane & 15;
    const int kOff4 = (lane >> 4) * 4;

    const float* __restrict__ rowA1 = f1  + (size_t)(waveRow + mA) * D_DIM + khalf * K_HALF + kOff4;
    const float* __restrict__ rowA2 = f2  + (size_t)(waveRow + mA) * D_DIM + khalf * K_HALF + kOff4;
    const float* __restrict__ rowB  = w1t + (size_t)nB * D_DIM             + khalf * K_HALF + kOff4;

    v8f c1 = {0.f, 0.f, 0.f, 0.f, 0.f, 0.f, 0.f, 0.f};
    v8f c2 = {0.f, 0.f, 0.f, 0.f, 0.f, 0.f, 0.f, 0.f};

    // ---- Gate GEMM: 8 K-columns per iteration = 3x b128 loads + 4 WMMAs ----
    #pragma unroll 4
    for (int k = 0; k < K_HALF; k += 8) {
        const float4 a1 = *(const float4*)(rowA1 + k);
        const float4 a2 = *(const float4*)(rowA2 + k);
        const float4 bb = *(const float4*)(rowB  + k);

        c1 = __builtin_amdgcn_wmma_f32_16x16x4_f32(false, lo2(a1), false, lo2(bb), (short)0, c1, false, false);
        c2 = __builtin_amdgcn_wmma_f32_16x16x4_f32(false, lo2(a2), false, lo2(bb), (short)0, c2, false, false);
        c1 = __builtin_amdgcn_wmma_f32_16x16x4_f32(false, hi2(a1), false, hi2(bb), (short)0, c1, false, false);
        c2 = __builtin_amdgcn_wmma_f32_16x16x4_f32(false, hi2(a2), false, hi2(bb), (short)0, c2, false, false);
    }

    // ---- Stage partial C fragments to LDS (striped -> row-major) ----
    if (nB < H_DIM) {
        const int mHi = (lane >> 4) * 8;
        #pragma unroll
        for (int r = 0; r < 8; ++r) {
            ldsC[wave][0][mHi + r][nB] = c1[r];
            ldsC[wave][1][mHi + r][nB] = c2[r];
        }
    }
    __syncthreads();

    // ---- 7-wide second layer + mean + 2-way softmax (1 thread per row) ----
    if (tid < ROWS_PER_BLOCK) {
        const int tp = tid >> 4;     // tile index
        const int m  = tid & 15;     // row within tile
        const int w0 = 2 * tp, w1w = 2 * tp + 1;

        float b2s = 0.f;
        #pragma unroll
        for (int o = 0; o < OUT_DIM; ++o) b2s += b2[o];

        float s1 = b2s, s2 = b2s;
        #pragma unroll
        for (int h = 0; h < H_DIM; ++h) {
            float w2s = 0.f;
            #pragma unroll
            for (int o = 0; o < OUT_DIM; ++o) w2s += W2[h * OUT_DIM + o];
            const float bb = b1[h];
            const float h1 = ldsC[w0][0][m][h] + ldsC[w1w][0][m][h] + bb;
            const float h2 = ldsC[w0][1][m][h] + ldsC[w1w][1][m][h] + bb;
            s1 += fmaxf(h1, 0.f) * w2s;
            s2 += fmaxf(h2, 0.f) * w2s;
        }
        const float l1 = s1 * (1.0f / 7.0f);
        const float l2 = s2 * (1.0f / 7.0f);
        const float a0 = 1.0f / (1.0f + __expf(l2 - l1));  // softmax_0 = sigmoid(l1-l2)
        ldsAlpha[tid][0] = a0;
        ldsAlpha[tid][1] = 1.0f - a0;
    }
    __syncthreads();

    // ---- Combine pass: out = a0*f1 + a1*f2 (tile re-reads hit L2) ----
    const int vecPerRow = D_DIM / 4;   // 1024 float4 per row
    for (int r = 0; r < ROWS_PER_BLOCK; ++r) {
        const float a0 = ldsAlpha[r][0];
        const float a1 = ldsAlpha[r][1];
        const size_t base = (size_t)(blockRow + r) * D_DIM;
        const float4* __restrict__ p1 = (const float4*)(f1 + base);
        const float4* __restrict__ p2 = (const float4*)(f2 + base);
        float4* __restrict__ po       = (float4*)(out + base);
        for (int c = tid; c < vecPerRow; c += BLOCK_THREADS) {
            const float4 x1 = p1[c];
            const float4 x2 = p2[c];
            float4 y;
            y.x = a0 * x1.x + a1 * x2.x;
            y.y = a0 * x1.y + a1 * x2.y;
            y.z = a0 * x1.z + a1 * x2.z;
            y.w = a0 * x1.w + a1 * x2.w;
            po[c] = y;
        }
    }
}

extern "C" void kernel_launch(void* const* d_in, const int* in_sizes, int n_in,
                              void* d_out, int out_size, void* d_ws, size_t ws_size,
                              hipStream_t stream) {
    const float* f1 = (const float*)d_in[0];
    const float* f2 = (const float*)d_in[1];
    const float* W1 = (const float*)d_in[2];
    const float* b1 = (const float*)d_in[3];
    const float* W2 = (const float*)d_in[4];
    const float* b2 = (const float*)d_in[5];
    float* out      = (float*)d_out;
    float* w1t      = (float*)d_ws;                 // needs 16*4096*4 = 256 KB

    // Build zero-padded W1^T in workspace (stream-ordered before main kernel).
    prep_w1t_kernel<<<dim3((N_PAD * D_DIM) / 256), 256, 0, stream>>>(W1, w1t);

    dim3 grid(B_ROWS / ROWS_PER_BLOCK);             // 256 blocks x 8 waves
    gating_fused_kernel<<<grid, BLOCK_THREADS, 0, stream>>>(
        f1, f2, w1t, b1, W2, b2, out);
}